// SciBERT_Graph_90993177133266
// MI455X (gfx1250) — compile-verified
//
#include <hip/hip_runtime.h>
#include <hip/hip_bf16.h>

typedef __attribute__((ext_vector_type(2))) float v2f;
typedef __attribute__((ext_vector_type(8))) float v8f;

// ---------------------------------------------------------------------------
// Workspace layout (floats):
//   agg       : N*8
//   deg       : N
//   relearned : N*8
// ---------------------------------------------------------------------------

__global__ void zero_ws_kernel(float* __restrict__ ws, int count) {
    int i = blockIdx.x * blockDim.x + threadIdx.x;
    int stride = gridDim.x * blockDim.x;
    for (; i < count; i += stride) ws[i] = 0.0f;
}

__global__ void edge_scatter_kernel(const float* __restrict__ node_emb,
                                    const int* __restrict__ esrc,
                                    const int* __restrict__ edst,
                                    float* __restrict__ agg,
                                    float* __restrict__ deg,
                                    int n_edges) {
    int e = blockIdx.x * blockDim.x + threadIdx.x;
    if (e >= n_edges) return;
    int s = esrc[e];
    int d = edst[e];
    const float4* row = (const float4*)(node_emb + (size_t)s * 8);
    float4 r0 = row[0];
    float4 r1 = row[1];
    float* a = agg + (size_t)d * 8;
    unsafeAtomicAdd(a + 0, r0.x);
    unsafeAtomicAdd(a + 1, r0.y);
    unsafeAtomicAdd(a + 2, r0.z);
    unsafeAtomicAdd(a + 3, r0.w);
    unsafeAtomicAdd(a + 4, r1.x);
    unsafeAtomicAdd(a + 5, r1.y);
    unsafeAtomicAdd(a + 6, r1.z);
    unsafeAtomicAdd(a + 7, r1.w);
    unsafeAtomicAdd(deg + d, 1.0f);
}

__global__ void sage_update_kernel(const float* __restrict__ node_emb,
                                   const float* __restrict__ agg,
                                   const float* __restrict__ deg,
                                   const float* __restrict__ W_self,
                                   const float* __restrict__ W_neigh,
                                   const float* __restrict__ sage_bias,
                                   float* __restrict__ relearned,
                                   int n_nodes) {
    __shared__ float sWs[64];
    __shared__ float sWn[64];
    __shared__ float sB[8];
    int t = threadIdx.x;
    if (t < 64) { sWs[t] = W_self[t]; sWn[t] = W_neigh[t]; }
    if (t < 8)  { sB[t] = sage_bias[t]; }
    __syncthreads();

    int i = blockIdx.x * blockDim.x + t;
    if (i >= n_nodes) return;

    const float4* nep = (const float4*)(node_emb + (size_t)i * 8);
    float4 n0 = nep[0], n1 = nep[1];
    float x[8] = {n0.x, n0.y, n0.z, n0.w, n1.x, n1.y, n1.z, n1.w};

    float dg = deg[i];
    float inv = 1.0f / fmaxf(dg, 1.0f);
    const float4* agp = (const float4*)(agg + (size_t)i * 8);
    float4 a0 = agp[0], a1 = agp[1];
    float h[8] = {a0.x * inv, a0.y * inv, a0.z * inv, a0.w * inv,
                  a1.x * inv, a1.y * inv, a1.z * inv, a1.w * inv};

    float out[8];
#pragma unroll
    for (int j = 0; j < 8; ++j) {
        float acc = sB[j];
#pragma unroll
        for (int k = 0; k < 8; ++k)
            acc += x[k] * sWs[j * 8 + k] + h[k] * sWn[j * 8 + k];
        out[j] = acc;
    }
    float4* rp = (float4*)(relearned + (size_t)i * 8);
    rp[0] = make_float4(out[0], out[1], out[2], out[3]);
    rp[1] = make_float4(out[4], out[5], out[6], out[7]);
}

// ---------------------------------------------------------------------------
// Fused MLP: per block, 16 batch rows. 8 waves of wave32.
//   stage 1: x1 = relu(bert[16,768] @ W1^T)          (8 waves x 192 f32-WMMA)
//   gather : x[:,128:144] = relearned[head/tail]
//   stage 2: x2 = relu(x[16,144] @ W2^T)             (wave 0, 36 f32-WMMA)
//   stage 3: out = x2 @ W3^T + b3                    (16 lanes, scalar)
// ---------------------------------------------------------------------------

#define XS_STR 772   // 768 + 4 pad  -> conflict-free ds_load_b64 A-fragments
#define X2_STR 148   // 144 + 4 pad

__global__ void __launch_bounds__(256)
fused_mlp_kernel(const float* __restrict__ bert,
                 const float* __restrict__ W1, const float* __restrict__ b1,
                 const float* __restrict__ W2, const float* __restrict__ b2,
                 const float* __restrict__ W3, const float* __restrict__ b3,
                 const float* __restrict__ relearned,
                 const int* __restrict__ row_ids,
                 float* __restrict__ out,
                 int batch) {
    __shared__ float xs[16 * XS_STR];   // 16x768 A tile (padded)
    __shared__ float x2[16 * X2_STR];   // 16x144 concat buffer (padded)
    __shared__ float x3[16 * 16];       // 16x16 hidden

    const int tid  = threadIdx.x;
    const int lane = tid & 31;
    const int wv   = tid >> 5;          // wave 0..7 -> N-tile
    const int nlo  = lane & 15;         // row (A) / col (B) index within tile
    const int kh   = (lane >> 4) << 1;  // K half: 0 or 2

    const size_t row0 = (size_t)blockIdx.x * 16;

    // ---- cooperative load of A tile (16x768 f32 = 48KB) ----
    const float4* asrc = (const float4*)(bert + row0 * 768);
    for (int c = tid; c < 16 * 192; c += 256) {
        int r = c / 192, q = c % 192;
        float4 v = asrc[r * 192 + q];
        *(float4*)&xs[r * XS_STR + q * 4] = v;
    }
    __syncthreads();

    // ---- stage 1: 16x16 tile of x1 per wave, K=768 via 16x16x4 f32 WMMA ----
    v8f c1 = {};
    const float* w1p = W1 + (size_t)(wv * 16 + nlo) * 768 + kh;
    const float* xap = &xs[nlo * XS_STR + kh];
#pragma unroll 4
    for (int k0 = 0; k0 < 768; k0 += 4) {
        v2f a = *(const v2f*)(xap + k0);
        v2f b = *(const v2f*)(w1p + k0);
        c1 = __builtin_amdgcn_wmma_f32_16x16x4_f32(
                 false, a, false, b, (short)0, c1, false, false);
    }
    {
        float bias = b1[wv * 16 + nlo];
        int mh = (lane >> 4) << 3;      // row offset 0 or 8
#pragma unroll
        for (int r = 0; r < 8; ++r) {
            float v = fmaxf(c1[r] + bias, 0.0f);
            x2[(r + mh) * X2_STR + wv * 16 + nlo] = v;
        }
    }

    // ---- gather head/tail relearned rows into cols 128..143 ----
    if (tid < 32) {
        int m = tid & 15, which = tid >> 4;
        int gm = (int)row0 + m;
        int idx = row_ids[which * batch + gm];
        const float4* src = (const float4*)(relearned + (size_t)idx * 8);
        float4 r0 = src[0], r1 = src[1];
        float* dst = &x2[m * X2_STR + 128 + which * 8];
        *(float4*)(dst + 0) = r0;
        *(float4*)(dst + 4) = r1;
    }
    __syncthreads();

    // ---- stage 2: x3 = relu(x[16,144] @ W2^T), wave 0 only (uniform branch) ----
    if (wv == 0) {
        v8f c2 = {};
        const float* w2p = W2 + (size_t)nlo * 144 + kh;
        const float* x2p = &x2[nlo * X2_STR + kh];
#pragma unroll
        for (int k0 = 0; k0 < 144; k0 += 4) {
            v2f a = *(const v2f*)(x2p + k0);
            v2f b = *(const v2f*)(w2p + k0);
            c2 = __builtin_amdgcn_wmma_f32_16x16x4_f32(
                     false, a, false, b, (short)0, c2, false, false);
        }
        float bias = b2[nlo];
        int mh = (lane >> 4) << 3;
#pragma unroll
        for (int r = 0; r < 8; ++r)
            x3[(r + mh) * 16 + nlo] = fmaxf(c2[r] + bias, 0.0f);
    }
    __syncthreads();

    // ---- stage 3: out[16,3] = x3 @ W3^T + b3 ----
    if (tid < 16) {
        float a0 = b3[0], a1 = b3[1], a2 = b3[2];
#pragma unroll
        for (int n = 0; n < 16; ++n) {
            float v = x3[tid * 16 + n];
            a0 += v * W3[n];
            a1 += v * W3[16 + n];
            a2 += v * W3[32 + n];
        }
        size_t gm = row0 + tid;
        out[gm * 3 + 0] = a0;
        out[gm * 3 + 1] = a1;
        out[gm * 3 + 2] = a2;
    }
}

// ---------------------------------------------------------------------------

extern "C" void kernel_launch(void* const* d_in, const int* in_sizes, int n_in,
                              void* d_out, int out_size, void* d_ws, size_t ws_size,
                              hipStream_t stream) {
    const float* bert      = (const float*)d_in[0];
    const float* node_emb  = (const float*)d_in[1];
    const int*   esrc      = (const int*)d_in[2];
    const int*   edst      = (const int*)d_in[3];
    const int*   row_ids   = (const int*)d_in[4];
    const float* W1        = (const float*)d_in[5];
    const float* b1        = (const float*)d_in[6];
    const float* W2        = (const float*)d_in[7];
    const float* b2        = (const float*)d_in[8];
    const float* W3        = (const float*)d_in[9];
    const float* b3        = (const float*)d_in[10];
    const float* W_self    = (const float*)d_in[11];
    const float* W_neigh   = (const float*)d_in[12];
    const float* sage_bias = (const float*)d_in[13];

    const int n_nodes = in_sizes[1] / 8;
    const int n_edges = in_sizes[2];
    const int batch   = in_sizes[4] / 2;

    float* agg       = (float*)d_ws;                 // N*8
    float* deg       = agg + (size_t)n_nodes * 8;    // N
    float* relearned = deg + (size_t)n_nodes;        // N*8

    // zero agg + deg (N*9 floats)
    {
        int count = n_nodes * 9;
        int blocks = (count + 255) / 256;
        if (blocks > 4096) blocks = 4096;
        zero_ws_kernel<<<blocks, 256, 0, stream>>>((float*)d_ws, count);
    }

    // scatter-add over edges
    edge_scatter_kernel<<<(n_edges + 255) / 256, 256, 0, stream>>>(
        node_emb, esrc, edst, agg, deg, n_edges);

    // per-node SAGE update
    sage_update_kernel<<<(n_nodes + 255) / 256, 256, 0, stream>>>(
        node_emb, agg, deg, W_self, W_neigh, sage_bias, relearned, n_nodes);

    // fused MLP over the batch, 16 rows per block
    fused_mlp_kernel<<<batch / 16, 256, 0, stream>>>(
        bert, W1, b1, W2, b2, W3, b3, relearned, row_ids,
        (float*)d_out, batch);
}